// Conv2d_NN_spatial_30743375905092
// MI455X (gfx1250) — compile-verified
//
#include <hip/hip_runtime.h>
#include <hip/hip_bf16.h>

// MI455X / gfx1250, wave32. Distance GEMM via V_WMMA_F32_16X16X4_F32 (fp32-exact),
// m^2 folded into an augmented K=68 contraction so WMMA output = score directly.
// Sample matrix staged with GLOBAL_LOAD_ASYNC_TO_LDS_B128 (ASYNCcnt path).
// Conv factored through samples: y[k][m][o] = W_k @ xs[:,m] (+bias in k=0), output
// is 3 gathered rows summed, written with fused pixel-shuffle.

typedef __attribute__((ext_vector_type(2))) float v2f;
typedef __attribute__((ext_vector_type(8))) float v8f;

#define BATCH 32
#define CIN 16
#define HIN 128
#define MSMP 256        // sample count
#define KDIM 68         // 64 feats + m2 + 3 pad (row stride; bank-conflict-free)
#define TOK_PER_WG 128
#define THREADS 256
#define NTOK 4096

__device__ __constant__ int c_sidx[16] = {0,4,8,13,17,21,25,29,34,38,42,46,50,55,59,63};

__device__ __forceinline__ unsigned lds_byte_off(const void* p) {
  // LDS flat addresses carry the LDS byte offset in the low 32 bits (ISA 10.2).
  return (unsigned)(unsigned long long)p;
}

// ---------------- Kernel 1: per-batch prep (xs rows + m2, y = W_k @ xs + bias) ----
__global__ void prep_kernel(const float* __restrict__ x, const float* __restrict__ w,
                            const float* __restrict__ bias, float* __restrict__ xsg,
                            float* __restrict__ yg) {
  const int b = blockIdx.x;
  const int tid = threadIdx.x;
  extern __shared__ float lds[];
  float* xsl = lds;                    // 256*68
  float* wl  = lds + MSMP * KDIM;      // 3*64*64, layout [k][c][o]
  float* bl  = wl + 3 * 64 * 64;       // 64

  for (int idx = tid; idx < 3 * 64 * 64; idx += THREADS) {
    int k = idx >> 12;                 // /4096
    int r = idx & 4095;
    int c = r >> 6, o = r & 63;
    wl[idx] = w[(o * 64 + c) * 3 + k]; // weight is [o][c][k]
  }
  if (tid < 64) bl[tid] = bias[tid];

  // gather sample m = tid (pixel-unshuffle indexing), compute m2
  const int m  = tid;
  const int h1 = c_sidx[m >> 4];
  const int w1 = c_sidx[m & 15];
  const float* xb = x + (size_t)b * CIN * HIN * HIN;
  float* xr = xsl + m * KDIM;
  float* xg = xsg + ((size_t)b * MSMP + m) * KDIM;
  float m2 = 0.f;
#pragma unroll
  for (int cc = 0; cc < 16; ++cc) {
    const float* p = xb + ((size_t)cc * HIN + 2 * h1) * HIN + 2 * w1;
    float2 l0 = *(const float2*)p;
    float2 l1 = *(const float2*)(p + HIN);
    float v0 = l0.x, v1 = l0.y, v2 = l1.x, v3 = l1.y;
    m2 += v0 * v0 + v1 * v1 + v2 * v2 + v3 * v3;
    xr[cc * 4 + 0] = v0; xr[cc * 4 + 1] = v1; xr[cc * 4 + 2] = v2; xr[cc * 4 + 3] = v3;
    xg[cc * 4 + 0] = v0; xg[cc * 4 + 1] = v1; xg[cc * 4 + 2] = v2; xg[cc * 4 + 3] = v3;
  }
  xr[64] = m2;
  xg[64] = m2; xg[65] = 0.f; xg[66] = 0.f; xg[67] = 0.f;
  __syncthreads();

  // y[k][m][o] = (k==0 ? bias[o] : 0) + sum_c W[o][c][k] * xs[c][m]
  const int o = tid & 63;
  const int g = tid >> 6;
  for (int item = g; item < 3 * MSMP; item += 4) {
    int k  = item >> 8;
    int mm = item & 255;
    float acc = (k == 0) ? bl[o] : 0.f;
    const float* wr = wl + k * 64 * 64 + o;
    const float* xv = xsl + mm * KDIM;
#pragma unroll 8
    for (int c = 0; c < 64; ++c) acc += wr[c * 64] * xv[c];
    yg[(((size_t)b * 3 + k) * MSMP + mm) * 64 + o] = acc;
  }
}

// ---------------- Kernel 2: WMMA distance scores + top-3 + gather/output ---------
// Branchless sorted-triple insert: straight-line v_cmp + v_cndmask, no exec churn.
__device__ __forceinline__ void top3_insert(float s, int m, float& s0, float& s1, float& s2,
                                            int& i0, int& i1, int& i2) {
  bool lt0 = s < s0, lt1 = s < s1, lt2 = s < s2;
  s2 = lt1 ? s1 : (lt2 ? s : s2);
  i2 = lt1 ? i1 : (lt2 ? m : i2);
  s1 = lt0 ? s0 : (lt1 ? s : s1);
  i1 = lt0 ? i0 : (lt1 ? m : i1);
  s0 = lt0 ? s  : s0;
  i0 = lt0 ? m  : i0;
}

__global__ void nn_conv_kernel(const float* __restrict__ x, const float* __restrict__ xsg,
                               const float* __restrict__ yg, float* __restrict__ out) {
  const int b  = blockIdx.x >> 5;
  const int n0 = (blockIdx.x & 31) * TOK_PER_WG;
  const int tid = threadIdx.x;
  extern __shared__ float lds[];
  float* axs = lds;                                   // 256*68 A-source (samples)
  float* btk = lds + MSMP * KDIM;                     // 128*68 B-source (tokens, -2x, 1)
  int*   idxs = (int*)(btk + TOK_PER_WG * KDIM);      // 128*3

  // ---- Phase A ----
  // samples: async DMA global->LDS (b128 per lane), tracked by ASYNCcnt
  {
    const char* src = (const char*)(xsg + (size_t)b * MSMP * KDIM);
    unsigned dst0 = lds_byte_off(axs) + (unsigned)tid * 16u;
    unsigned off0 = (unsigned)tid * 16u;
#pragma unroll
    for (int i = 0; i < (MSMP * KDIM * 4) / (THREADS * 16); ++i) {  // 17
      unsigned d = dst0 + (unsigned)(i * THREADS * 16);
      unsigned v = off0 + (unsigned)(i * THREADS * 16);
      asm volatile("global_load_async_to_lds_b128 %0, %1, %2 offset:0"
                   :: "v"(d), "v"(v), "s"(src)
                   : "memory");
    }
  }
  // tokens: pixel-unshuffle gather with -2x scaling (needs VALU, normal path)
  {
    const float* xb = x + (size_t)b * CIN * HIN * HIN;
#pragma unroll
    for (int i = 0; i < (TOK_PER_WG * 16) / THREADS; ++i) {         // 8
      int task = tid + i * THREADS;
      int t  = task & (TOK_PER_WG - 1);
      int cc = task >> 7;
      int n  = n0 + t;
      int h1 = n >> 6, w1 = n & 63;
      const float* p = xb + ((size_t)cc * HIN + 2 * h1) * HIN + 2 * w1;
      float2 l0 = *(const float2*)p;
      float2 l1 = *(const float2*)(p + HIN);
      float* q = btk + t * KDIM + cc * 4;
      q[0] = -2.f * l0.x; q[1] = -2.f * l0.y; q[2] = -2.f * l1.x; q[3] = -2.f * l1.y;
    }
    if (tid < TOK_PER_WG) {
      float* q = btk + tid * KDIM + 64;
      q[0] = 1.f; q[1] = 0.f; q[2] = 0.f; q[3] = 0.f;
    }
  }
  asm volatile("s_wait_asynccnt 0" ::: "memory");
  __syncthreads();

  // ---- Phase B: score GEMM via wmma_f32_16x16x4_f32, K=68 (17 steps) ----
  const int lane = tid & 31;
  const int wv   = tid >> 5;       // wave owns 16 tokens
  const int tl   = lane & 15;
  const int hi   = lane >> 4;

  v2f bf[17];
  {
    const float* br = btk + (wv * 16 + tl) * KDIM + 2 * hi;
#pragma unroll
    for (int kk = 0; kk < 17; ++kk) bf[kk] = *(const v2f*)(br + kk * 4);
  }

  float s0 = 3.4e38f, s1 = 3.4e38f, s2 = 3.4e38f;
  int   i0 = 0, i1 = 0, i2 = 0;

  // 2 sample-tiles per iteration: two independent WMMA accumulator chains (ILP)
  for (int mt = 0; mt < 16; mt += 2) {
    v8f acc0 = {0.f, 0.f, 0.f, 0.f, 0.f, 0.f, 0.f, 0.f};
    v8f acc1 = {0.f, 0.f, 0.f, 0.f, 0.f, 0.f, 0.f, 0.f};
    const float* ar0 = axs + (mt * 16 + tl) * KDIM + 2 * hi;
    const float* ar1 = ar0 + 16 * KDIM;
#pragma unroll
    for (int kk = 0; kk < 17; ++kk) {
      v2f a0 = *(const v2f*)(ar0 + kk * 4);
      v2f a1 = *(const v2f*)(ar1 + kk * 4);
      acc0 = __builtin_amdgcn_wmma_f32_16x16x4_f32(false, a0, false, bf[kk],
                                                   (short)0, acc0, false, false);
      acc1 = __builtin_amdgcn_wmma_f32_16x16x4_f32(false, a1, false, bf[kk],
                                                   (short)0, acc1, false, false);
    }
    const int mb0 = mt * 16 + hi * 8;        // D: VGPR j -> sample mb+j
    const int mb1 = mb0 + 16;
#pragma unroll
    for (int j = 0; j < 8; ++j)
      top3_insert(acc0[j], mb0 + j, s0, s1, s2, i0, i1, i2);
#pragma unroll
    for (int j = 0; j < 8; ++j)
      top3_insert(acc1[j], mb1 + j, s0, s1, s2, i0, i1, i2);
  }

  // merge across half-wave (lane^16 saw the other 8 samples of every tile)
  {
    float ps0 = __shfl_xor(s0, 16), ps1 = __shfl_xor(s1, 16), ps2 = __shfl_xor(s2, 16);
    int   pj0 = __shfl_xor(i0, 16), pj1 = __shfl_xor(i1, 16), pj2 = __shfl_xor(i2, 16);
    top3_insert(ps0, pj0, s0, s1, s2, i0, i1, i2);
    top3_insert(ps1, pj1, s0, s1, s2, i0, i1, i2);
    top3_insert(ps2, pj2, s0, s1, s2, i0, i1, i2);
  }
  if (hi == 0) {
    int* q = idxs + (wv * 16 + tl) * 3;
    q[0] = i0; q[1] = i1; q[2] = i2;
  }
  __syncthreads();

  // ---- Phase C: gather y rows, sum, write with fused pixel-shuffle ----
  const int o = tid & 63;
  const int g = tid >> 6;
  const float* yb = yg + (size_t)b * 3 * MSMP * 64;
#pragma unroll 4
  for (int i = 0; i < TOK_PER_WG / 4; ++i) {
    int t  = g + i * 4;
    int j0 = idxs[t * 3 + 0], j1 = idxs[t * 3 + 1], j2 = idxs[t * 3 + 2];
    float v = yb[(0 * MSMP + j0) * 64 + o]
            + yb[(1 * MSMP + j1) * 64 + o]
            + yb[(2 * MSMP + j2) * 64 + o];
    int n = n0 + t;
    int h1 = n >> 6, w1 = n & 63;
    out[(((size_t)b * CIN + (o >> 2)) * HIN + 2 * h1 + ((o >> 1) & 1)) * HIN
        + 2 * w1 + (o & 1)] = v;
  }
}

extern "C" void kernel_launch(void* const* d_in, const int* in_sizes, int n_in,
                              void* d_out, int out_size, void* d_ws, size_t ws_size,
                              hipStream_t stream) {
  const float* x    = (const float*)d_in[0];
  const float* w    = (const float*)d_in[1];
  const float* bias = (const float*)d_in[2];
  float* out = (float*)d_out;

  // workspace: xsg [32][256][68] f32 (2.23 MB) then yg [32][3][256][64] f32 (6.29 MB)
  float* xsg = (float*)d_ws;
  float* yg  = xsg + (size_t)BATCH * MSMP * KDIM;

  size_t lds1 = (size_t)(MSMP * KDIM + 3 * 64 * 64 + 64) * sizeof(float);                 // ~116 KB
  size_t lds2 = (size_t)(MSMP * KDIM + TOK_PER_WG * KDIM) * sizeof(float)
              + (size_t)TOK_PER_WG * 3 * sizeof(int);                                     // ~104 KB

  prep_kernel<<<BATCH, THREADS, lds1, stream>>>(x, w, bias, xsg, yg);
  nn_conv_kernel<<<BATCH * (NTOK / TOK_PER_WG), THREADS, lds2, stream>>>(x, xsg, yg, out);
}